// LTC_5634997093285
// MI455X (gfx1250) — compile-verified
//
#include <hip/hip_runtime.h>
#include <stdint.h>

// LTC cell, B=16 T=128 C=64 U=128, 6 ODE unfolds.
// One block per batch element; 1024 threads = 32 wave32 waves on one WGP.
// seg = tid>>7 splits pre-neuron dim 8-way; j = tid&127 is the post neuron.

#define B_ 16
#define T_ 128
#define C_ 64
#define U_ 128
#define UNFOLDS_ 6

__device__ __forceinline__ float softplus_f(float x) {
    return (x > 20.0f) ? x : logf(1.0f + expf(x));
}

// den is strictly positive here (cm_t + gleak + nonneg sums + eps), so a
// v_rcp_f32 + one Newton step (~1 ulp) replaces the 10-op IEEE div sequence
// on the 768-deep dependent chain.
__device__ __forceinline__ float fast_div_pos(float num, float den) {
    float r = __builtin_amdgcn_rcpf(den);
    r = r * __builtin_fmaf(-den, r, 2.0f);   // Newton-Raphson refine
    return num * r;
}

__launch_bounds__(1024, 1)
__global__ void ltc_kernel(const float* __restrict__ x,
                           const float* __restrict__ h0,
                           const float* __restrict__ gleak,
                           const float* __restrict__ vleak,
                           const float* __restrict__ cm,
                           const float* __restrict__ sigma,
                           const float* __restrict__ mu,
                           const float* __restrict__ w,
                           const float* __restrict__ erev,
                           const float* __restrict__ ssigma,
                           const float* __restrict__ smu,
                           const float* __restrict__ sw,
                           const float* __restrict__ serev,
                           const float* __restrict__ iw,
                           const float* __restrict__ ib,
                           const float* __restrict__ ow,
                           const float* __restrict__ ob,
                           float* __restrict__ out)
{
    const int tid = (int)threadIdx.x;
    const int j   = tid & (U_ - 1);   // post neuron
    const int seg = tid >> 7;         // 0..7, pre-dim slice
    const int b   = (int)blockIdx.x;  // batch element

    __shared__ float vlds[U_];
    __shared__ float pnum[8][U_];     // recurrent partials
    __shared__ float pden[8][U_];
    __shared__ float ssn[8][U_];      // sensory partials
    __shared__ float ssd[8][U_];
    __shared__ float xbuf[2][C_];     // double-buffered x[b,t,:]

    const float LOG2E = 1.44269504088896340736f;

    // ---- one-time parameter transform into registers ----
    // recurrent: arg = S[k]*v + Cc[k]  with  sigmoid(z)=rcp(1+exp2(-z*log2e))
    float S[16], Cc[16], W[16];
    #pragma unroll
    for (int k = 0; k < 16; ++k) {
        const int i   = seg * 16 + k;
        const int idx = i * U_ + j;
        const float sg = sigma[idx];
        const float m  = mu[idx];
        const float wp = softplus_f(w[idx]);
        S[k]  = -sg * LOG2E;
        Cc[k] =  sg * m * LOG2E;
        W[k]  =  wp * erev[idx];      // |W[k]| recovered by free src-abs modifier
    }
    // sensory with input affine folded in: arg = SA[k]*x_c + SB[k]
    float SA[8], SB[8], SW[8];
    #pragma unroll
    for (int k = 0; k < 8; ++k) {
        const int c   = seg * 8 + k;
        const int idx = c * U_ + j;
        const float sg = ssigma[idx];
        const float m  = smu[idx];
        const float wp = softplus_f(sw[idx]);
        SA[k] = -sg * LOG2E * iw[c];
        SB[k] =  sg * LOG2E * (m - ib[c]);
        SW[k] =  wp * serev[idx];
    }
    const float cmt  = softplus_f(cm[j]) * (float)UNFOLDS_;
    const float gl   = softplus_f(gleak[j]);
    const float glvl = gl * vleak[j];
    const float denc = cmt + gl + 1e-8f;
    const float owj  = ow[j];
    const float obj  = ob[j];

    // ---- init state and first x slice ----
    if (seg == 0) vlds[j] = h0[b * U_ + j];
    if (tid < C_) xbuf[0][tid] = x[((size_t)b * T_ + 0) * C_ + tid];
    __syncthreads();

    float vj = (seg == 0) ? vlds[j] : 0.0f;  // seg0 owns the live state for its j

    for (int t = 0; t < T_; ++t) {
        const int cur = t & 1;

        // prefetch (issued last iteration) must have landed in LDS before reads
        asm volatile("s_wait_asynccnt 0" ::: "memory");
        __syncthreads();

        // async prefetch of x[b,t+1,:] into the other buffer (16 lanes x 16B)
        if ((t + 1 < T_) && (tid < 16)) {
            uint32_t ldsa = (uint32_t)(uintptr_t)(&xbuf[cur ^ 1][tid * 4]);
            uint64_t ga   = (uint64_t)(const void*)(x + ((size_t)b * T_ + (t + 1)) * C_ + tid * 4);
            asm volatile("global_load_async_to_lds_b128 %0, %1, off"
                         :: "v"(ldsa), "v"(ga) : "memory");
        }

        // ---- sensory synapses (constant across unfolds) ----
        {
            float sn = 0.0f, sd = 0.0f;
            #pragma unroll
            for (int k = 0; k < 8; ++k) {
                const float xc  = xbuf[cur][seg * 8 + k];
                const float arg = __builtin_fmaf(SA[k], xc, SB[k]);
                const float e   = __builtin_amdgcn_exp2f(arg);
                const float r   = __builtin_amdgcn_rcpf(1.0f + e);
                sn = __builtin_fmaf(SW[k], r, sn);
                sd = __builtin_fmaf(__builtin_fabsf(SW[k]), r, sd);
            }
            ssn[seg][j] = sn;
            ssd[seg][j] = sd;
            // no barrier needed yet: first unfold's first barrier orders these
        }

        float snum = 0.0f, sden = 0.0f;  // seg0 accumulates once at unfold 0

        // ---- 6 ODE unfolds ----
        #pragma unroll 1
        for (int u = 0; u < UNFOLDS_; ++u) {
            float rn = 0.0f, rd = 0.0f;
            #pragma unroll
            for (int k = 0; k < 16; ++k) {
                const float vi  = vlds[seg * 16 + k];
                const float arg = __builtin_fmaf(S[k], vi, Cc[k]);
                const float e   = __builtin_amdgcn_exp2f(arg);
                const float r   = __builtin_amdgcn_rcpf(1.0f + e);
                rn = __builtin_fmaf(W[k], r, rn);
                rd = __builtin_fmaf(__builtin_fabsf(W[k]), r, rd);
            }
            pnum[seg][j] = rn;
            pden[seg][j] = rd;
            __syncthreads();   // partials (and at u==0, sensory partials) visible

            if (seg == 0) {
                if (u == 0) {
                    #pragma unroll
                    for (int s = 0; s < 8; ++s) { snum += ssn[s][j]; sden += ssd[s][j]; }
                }
                float tn = snum, td = sden;
                #pragma unroll
                for (int s = 0; s < 8; ++s) { tn += pnum[s][j]; td += pden[s][j]; }
                const float num = __builtin_fmaf(cmt, vj, glvl) + tn;
                const float den = denc + td;
                vj = fast_div_pos(num, den);
                vlds[j] = vj;
            }
            __syncthreads();   // v visible; pnum/pden safe to overwrite
        }

        // readout for this timestep
        if (seg == 0) {
            out[((size_t)b * T_ + t) * U_ + j] = __builtin_fmaf(vj, owj, obj);
        }
    }

    // final hidden state after the readout block
    if (seg == 0) {
        out[(size_t)B_ * T_ * U_ + (size_t)b * U_ + j] = vj;
    }
}

extern "C" void kernel_launch(void* const* d_in, const int* in_sizes, int n_in,
                              void* d_out, int out_size, void* d_ws, size_t ws_size,
                              hipStream_t stream) {
    (void)in_sizes; (void)n_in; (void)out_size; (void)d_ws; (void)ws_size;
    const float* x      = (const float*)d_in[0];
    const float* h0     = (const float*)d_in[1];
    const float* gleak  = (const float*)d_in[2];
    const float* vleak  = (const float*)d_in[3];
    const float* cm     = (const float*)d_in[4];
    const float* sigma  = (const float*)d_in[5];
    const float* mu     = (const float*)d_in[6];
    const float* w      = (const float*)d_in[7];
    const float* erev   = (const float*)d_in[8];
    const float* ssig   = (const float*)d_in[9];
    const float* smu    = (const float*)d_in[10];
    const float* sw     = (const float*)d_in[11];
    const float* serev  = (const float*)d_in[12];
    const float* iw     = (const float*)d_in[13];
    const float* ibv    = (const float*)d_in[14];
    const float* ow     = (const float*)d_in[15];
    const float* ob     = (const float*)d_in[16];
    float* out = (float*)d_out;

    ltc_kernel<<<dim3(B_), dim3(1024), 0, stream>>>(
        x, h0, gleak, vleak, cm, sigma, mu, w, erev,
        ssig, smu, sw, serev, iw, ibv, ow, ob, out);
}